// GCNDiscriminator_53326313947142
// MI455X (gfx1250) — compile-verified
//
#include <hip/hip_runtime.h>

// GCN discriminator, fused per-graph kernel for gfx1250 (MI455X).
// Core GEMM (A[512x512] @ P[512x64], fp32) runs on V_WMMA_F32_16X16X4_F32.
// P is stored k-pair-interleaved in LDS so each WMMA B-fragment is one
// contiguous ds_load_b64 into an even VGPR pair (no repacking moves).

typedef float v2f __attribute__((ext_vector_type(2)));
typedef float v8f __attribute__((ext_vector_type(8)));

#define NB      512   // batch (graphs)
#define NN      512   // nodes
#define HIDN    64
#define PP_STR  160   // floats per k-pair row: 128 used + 32 pad (mod 64 == 32
                      // -> the two lane-halves of a wave hit disjoint LDS banks)
#define THREADS 256   // 8 waves (wave32)

static constexpr unsigned SMEM_FLOATS =
    (NN / 2) * PP_STR   // Pp_s: P[k][h] at Pp[k>>1][h*2 + (k&1)]
  + NN                  // d_s
  + NN                  // r_s
  + HIDN * HIDN         // W2_s
  + HIDN                // w1_s
  + HIDN                // b1_s
  + HIDN                // b2_s
  + HIDN;               // gp_s (pool accumulator)
static constexpr unsigned SMEM_BYTES = SMEM_FLOATS * sizeof(float);

__device__ __forceinline__ int pp_idx(int k, int h) {
    return (k >> 1) * PP_STR + (h << 1) + (k & 1);
}

__launch_bounds__(THREADS, 1)
__global__ void gcn_fused(const float* __restrict__ A,
                          const float* __restrict__ W1, const float* __restrict__ b1,
                          const float* __restrict__ W2, const float* __restrict__ b2,
                          const float* __restrict__ Wl, const float* __restrict__ bl,
                          float* __restrict__ out)
{
    extern __shared__ float smem[];
    float* Pp_s = smem;                       // [(NN/2)][PP_STR]
    float* d_s  = Pp_s + (NN / 2) * PP_STR;   // [NN]
    float* r_s  = d_s  + NN;                  // [NN]
    float* W2_s = r_s  + NN;                  // [HIDN][HIDN]
    float* w1_s = W2_s + HIDN * HIDN;         // [HIDN]
    float* b1_s = w1_s + HIDN;                // [HIDN]
    float* b2_s = b1_s + HIDN;                // [HIDN]
    float* gp_s = b2_s + HIDN;                // [HIDN]

    const int tid = threadIdx.x;
    const int b   = blockIdx.x;
    const float* __restrict__ Ab = A + (size_t)b * NN * NN;

    // ---- stage small weights, zero pool accumulator
    for (int i = tid; i < HIDN * HIDN; i += THREADS) W2_s[i] = W2[i];
    if (tid < HIDN) {
        w1_s[tid] = W1[tid];
        b1_s[tid] = b1[tid];
        b2_s[tid] = b2[tid];
        gp_s[tid] = 0.0f;
    }

    // ---- phase 1: degrees. A is symmetric -> row sums == column sums,
    //      accumulate column-wise for perfectly coalesced loads.
    const int c0 = tid, c1 = tid + THREADS;
    float s0 = 0.0f, s1 = 0.0f;
    for (int j = 0; j < NN; ++j) {
        s0 += Ab[(size_t)j * NN + c0];
        s1 += Ab[(size_t)j * NN + c1];
    }
    d_s[c0] = 1.0f / sqrtf(1.0f + s0);   // deg+1 >= 1, finite
    d_s[c1] = 1.0f / sqrtf(1.0f + s1);
    __syncthreads();

    // ---- phase 2: Av = A @ d (symmetry again -> column-wise accumulation)
    float a0 = 0.0f, a1 = 0.0f;
    for (int j = 0; j < NN; ++j) {
        const float dj = d_s[j];
        a0 = fmaf(Ab[(size_t)j * NN + c0], dj, a0);
        a1 = fmaf(Ab[(size_t)j * NN + c1], dj, a1);
    }
    r_s[c0] = d_s[c0] * (a0 + d_s[c0]);  // r_i = sum_j An[i,j]
    r_s[c1] = d_s[c1] * (a1 + d_s[c1]);
    __syncthreads();

    // ---- phase 3: P[i][h] = d_i * ( relu(r_i*W1 + b1) @ W2 )[h]
    for (int task = tid; task < NN * 4; task += THREADS) {
        const int row = task >> 2;
        const int seg = (task & 3) << 4;       // 16-wide output segment
        const float r  = r_s[row];
        const float dd = d_s[row];
        float acc[16];
        #pragma unroll
        for (int j = 0; j < 16; ++j) acc[j] = 0.0f;
        for (int hin = 0; hin < HIDN; ++hin) {
            const float h1 = fmaxf(fmaf(r, w1_s[hin], b1_s[hin]), 0.0f);
            const float* w2r = W2_s + hin * HIDN + seg;
            #pragma unroll
            for (int j = 0; j < 16; ++j) acc[j] = fmaf(h1, w2r[j], acc[j]);
        }
        #pragma unroll
        for (int j = 0; j < 16; ++j) Pp_s[pp_idx(row, seg + j)] = dd * acc[j];
    }
    __syncthreads();

    // ---- phase 4: core = A @ P via V_WMMA_F32_16X16X4_F32, fused epilogue.
    // A-frag 16x4 f32 layout: lanes 0-15 rows, v0 = K{0|2}, v1 = K{1|3} by lane half.
    // B-frag 4x16 f32 mirrors A: one contiguous float2 in the pair layout.
    // D 16x16: VGPR r holds row (r + 8*(lane>=16)), col = lane%16.
    const int wave = tid >> 5;
    const int lane = tid & 31;
    const int lm   = lane & 15;
    const int lh   = lane >> 4;   // 0 | 1
    for (int mt = wave; mt < NN / 16; mt += THREADS / 32) {
        const int m0 = mt * 16;
        v8f acc[4];
        #pragma unroll
        for (int n = 0; n < 4; ++n)
            #pragma unroll
            for (int e = 0; e < 8; ++e) acc[n][e] = 0.0f;

        const float* arow = Ab + (size_t)(m0 + lm) * NN + (lh << 1);
        for (int k0 = 0; k0 < NN; k0 += 4) {
            const v2f av = *reinterpret_cast<const v2f*>(arow + k0);  // 8B, aligned
            // k-pair row for this lane half: (k0 + 2*lh)/2 = k0/2 + lh
            const float* bbase = Pp_s + ((k0 >> 1) + lh) * PP_STR;
            #pragma unroll
            for (int n = 0; n < 4; ++n) {
                const int col = (n << 4) + lm;
                const v2f bv = *reinterpret_cast<const v2f*>(bbase + (col << 1)); // ds_load_b64
                acc[n] = __builtin_amdgcn_wmma_f32_16x16x4_f32(
                    /*neg_a=*/false, av, /*neg_b=*/false, bv,
                    /*c_mod=*/(short)0, acc[n],
                    /*reuse_a=*/false, /*reuse_b=*/false);
            }
        }

        // epilogue: h2 = relu(d_i*(core + P) + b2); accumulate column sums (pool)
        #pragma unroll
        for (int n = 0; n < 4; ++n) {
            const int col = (n << 4) + lm;
            const float b2v = b2_s[col];
            float csum = 0.0f;
            #pragma unroll
            for (int rr = 0; rr < 8; ++rr) {
                const int row = m0 + rr + (lh << 3);
                const float v = fmaf(d_s[row], acc[n][rr] + Pp_s[pp_idx(row, col)], b2v);
                csum += fmaxf(v, 0.0f);
            }
            atomicAdd(&gp_s[col], csum);   // ds_add_f32
        }
    }
    __syncthreads();

    // ---- phase 5: logits[b] = mean_pool @ Wl + bl
    if (tid == 0) {
        float s = 0.0f;
        for (int h = 0; h < HIDN; ++h) s = fmaf(gp_s[h], Wl[h], s);
        out[b] = s * (1.0f / NN) + bl[0];
    }
}

extern "C" void kernel_launch(void* const* d_in, const int* in_sizes, int n_in,
                              void* d_out, int out_size, void* d_ws, size_t ws_size,
                              hipStream_t stream) {
    (void)in_sizes; (void)n_in; (void)d_ws; (void)ws_size; (void)out_size;
    const float* A  = (const float*)d_in[0];
    const float* W1 = (const float*)d_in[1];
    const float* b1 = (const float*)d_in[2];
    const float* W2 = (const float*)d_in[3];
    const float* b2 = (const float*)d_in[4];
    const float* Wl = (const float*)d_in[5];
    const float* bl = (const float*)d_in[6];
    float* out = (float*)d_out;

    // ~181 KB LDS per workgroup (CDNA5 WGP has 320 KB) -> dynamic shared mem.
    hipFuncSetAttribute((const void*)gcn_fused,
                        hipFuncAttributeMaxDynamicSharedMemorySize,
                        (int)SMEM_BYTES);
    gcn_fused<<<NB, THREADS, SMEM_BYTES, stream>>>(A, W1, b1, W2, b2, Wl, bl, out);
}